// ProgressiveWaveContextEncoder_7859790151685
// MI455X (gfx1250) — compile-verified
//
#include <hip/hip_runtime.h>

typedef __attribute__((ext_vector_type(2))) float v2f;
typedef __attribute__((ext_vector_type(8))) float v8f;

#define B_TOT   512
#define T_PAST  360
#define T_FUTS  1800
#define WSPF    34
#define FUT     50
#define T_TOT   (T_PAST + FUT * WSPF)   /* 2060 */
#define H       64
#define HS      68                      /* hbuf row stride (pad: 64-bank conflict free) */
#define GS      17                      /* gate tile row stride */

__device__ __forceinline__ float rcp_f(float x) { return __builtin_amdgcn_rcpf(x); }
__device__ __forceinline__ float sigm_f(float x) { return rcp_f(1.0f + __expf(-x)); }
__device__ __forceinline__ float tanh_f(float x) { return 1.0f - 2.0f * rcp_f(1.0f + __expf(2.0f * x)); }

// Persistent LSTM scan: one block = 16 batch rows, 16 waves = 16 gate-column tiles.
__global__ __launch_bounds__(512) void lstm_scan_kernel(
    const float* __restrict__ xin,   // [512, 360, 3]
    const float* __restrict__ xfut,  // [512, 1800, 3]
    const float* __restrict__ W_ih,  // [256, 3]
    const float* __restrict__ W_hh,  // [256, 64]
    const float* __restrict__ b_ih,  // [256]
    const float* __restrict__ b_hh,  // [256]
    float* __restrict__ hsel)        // == d_out, [512, 50, 64]
{
    __shared__ float hbuf[16 * HS];       // h_t for this batch group, padded
    __shared__ float xbuf[16 * 4];        // x_t for this batch group
    __shared__ float gbuf[12 * 16 * GS];  // f,g,o gate tiles from waves 4..15

    const int tid  = threadIdx.x;
    const int lane = tid & 31;
    const int wave = tid >> 5;        // 0..15
    const int hi   = lane >> 4;       // half-wave select
    const int ml   = lane & 15;       // m (for A/copy) or n-local (for B/C/D)
    const int base = blockIdx.x * 16; // batch group base

    const int n = wave * 16 + ml;     // global gate row 0..255 (i|f|g|o blocks of 64)

    // Preload this wave's W_hh^T tile slices into registers (B operand, 16 k-slabs).
    v2f bw[16];
#pragma unroll
    for (int kt = 0; kt < 16; ++kt) {
        const int k = kt * 4 + hi * 2;
        bw[kt][0] = W_hh[n * H + k];
        bw[kt][1] = W_hh[n * H + k + 1];
    }
    const float wi0 = W_ih[n * 3 + 0];
    const float wi1 = W_ih[n * 3 + 1];
    const float wi2 = W_ih[n * 3 + 2];
    const float bias = b_ih[n] + b_hh[n];

    // h0 = 0; stage x[0]
    for (int i = tid; i < 16 * HS; i += 512) hbuf[i] = 0.0f;
    if (wave == 0 && lane < 16) {
        const float* p = xin + (size_t)(base + lane) * T_PAST * 3;
        xbuf[lane * 4 + 0] = p[0];
        xbuf[lane * 4 + 1] = p[1];
        xbuf[lane * 4 + 2] = p[2];
    }
    __syncthreads();

    v8f c = {};  // cell state (used by waves 0..3 only)

    for (int t = 0; t < T_TOT; ++t) {
        // Two independent accumulator chains -> 2-way ILP on the XDL pipe.
        // acc0 starts from bias + x_t @ W_ih^T; acc1 starts from zero.
        v8f acc0, acc1 = {};
#pragma unroll
        for (int r = 0; r < 8; ++r) {
            const int m = r + 8 * hi;
            acc0[r] = bias + wi0 * xbuf[m * 4 + 0] + wi1 * xbuf[m * 4 + 1] + wi2 * xbuf[m * 4 + 2];
        }
        // gates += h_t @ W_hh^T via fp32 WMMA, K=64 in 16 slabs of 4 (even/odd split)
#pragma unroll
        for (int kt = 0; kt < 16; kt += 2) {
            const int kb0 = kt * 4 + hi * 2;
            const int kb1 = (kt + 1) * 4 + hi * 2;
            v2f a0, a1;
            a0[0] = hbuf[ml * HS + kb0];
            a0[1] = hbuf[ml * HS + kb0 + 1];
            a1[0] = hbuf[ml * HS + kb1];
            a1[1] = hbuf[ml * HS + kb1 + 1];
            acc0 = __builtin_amdgcn_wmma_f32_16x16x4_f32(
                false, a0, false, bw[kt], (short)0, acc0, false, false);
            acc1 = __builtin_amdgcn_wmma_f32_16x16x4_f32(
                false, a1, false, bw[kt + 1], (short)0, acc1, false, false);
        }
        v8f acc;
#pragma unroll
        for (int r = 0; r < 8; ++r) acc[r] = acc0[r] + acc1[r];

        // Prefetch next step's x while gates settle (wave 0 only).
        float nx0 = 0.0f, nx1 = 0.0f, nx2 = 0.0f;
        if (wave == 0 && lane < 16 && (t + 1) < T_TOT) {
            const int tn = t + 1;
            const float* p = (tn < T_PAST)
                ? (xin  + ((size_t)(base + lane) * T_PAST + tn) * 3)
                : (xfut + ((size_t)(base + lane) * T_FUTS + (tn - T_PAST)) * 3);
            nx0 = p[0]; nx1 = p[1]; nx2 = p[2];
        }

        // Waves 4..15 publish f/g/o tiles.
        if (wave >= 4) {
            const int slot = wave - 4;
#pragma unroll
            for (int r = 0; r < 8; ++r) {
                const int m = r + 8 * hi;
                gbuf[slot * 16 * GS + m * GS + ml] = acc[r];
            }
        }
        __syncthreads();  // gate tiles visible; all hbuf/xbuf reads done

        // Waves 0..3 own columns [wave*16, wave*16+16) of c/h.
        if (wave < 4) {
#pragma unroll
            for (int r = 0; r < 8; ++r) {
                const int m  = r + 8 * hi;
                const float iv = sigm_f(acc[r]);
                const float fv = sigm_f(gbuf[(wave)     * 16 * GS + m * GS + ml]);
                const float gv = tanh_f(gbuf[(4 + wave) * 16 * GS + m * GS + ml]);
                const float ov = sigm_f(gbuf[(8 + wave) * 16 * GS + m * GS + ml]);
                const float cv = fv * c[r] + iv * gv;
                c[r] = cv;
                hbuf[m * HS + wave * 16 + ml] = ov * tanh_f(cv);
            }
        }
        if (wave == 0 && lane < 16 && (t + 1) < T_TOT) {
            xbuf[lane * 4 + 0] = nx0;
            xbuf[lane * 4 + 1] = nx1;
            xbuf[lane * 4 + 2] = nx2;
        }
        __syncthreads();  // h_{t+1}, x_{t+1} published

        // Selected timesteps: t = 360 + 34*(j+1) - 1 -> copy h to d_out rows.
        if (t >= T_PAST + WSPF - 1 && ((t - (T_PAST + WSPF - 1)) % WSPF) == 0) {
            const int j = (t - (T_PAST + WSPF - 1)) / WSPF;  // 0..49
            const int m = tid >> 5;         // 0..15
            const int k = (tid & 31) * 2;   // 0..62 step 2
            const float2 hv = *(const float2*)&hbuf[m * HS + k];
            *(float2*)(hsel + ((size_t)(base + m) * FUT + j) * H + k) = hv;
        }
    }
}

// In-place projection on d_out: row = (b*50 + f), out[row,:] = out[row,:] @ W_proj^T + b_proj.
__global__ __launch_bounds__(128) void proj_kernel(
    const float* __restrict__ W_proj,  // [64, 64]
    const float* __restrict__ b_proj,  // [64]
    float* __restrict__ out)           // [25600, 64], read+write
{
    const int tid  = threadIdx.x;
    const int lane = tid & 31;
    const int wave = tid >> 5;    // 0..3 -> output cols [wave*16, wave*16+16)
    const int hi   = lane >> 4;
    const int ml   = lane & 15;
    const int n    = wave * 16 + ml;
    const size_t rb = (size_t)blockIdx.x * 16;

    v2f bw[16];
#pragma unroll
    for (int kt = 0; kt < 16; ++kt) {
        const int k = kt * 4 + hi * 2;
        bw[kt][0] = W_proj[n * H + k];
        bw[kt][1] = W_proj[n * H + k + 1];
    }
    const float bp = b_proj[n];

    v8f acc0, acc1 = {};
#pragma unroll
    for (int r = 0; r < 8; ++r) acc0[r] = bp;

    v2f a[16];
#pragma unroll
    for (int kt = 0; kt < 16; ++kt) {
        const int k = kt * 4 + hi * 2;
        const float* p = out + (rb + ml) * H + k;
        a[kt][0] = p[0];
        a[kt][1] = p[1];
    }
#pragma unroll
    for (int kt = 0; kt < 16; kt += 2) {
        acc0 = __builtin_amdgcn_wmma_f32_16x16x4_f32(
            false, a[kt], false, bw[kt], (short)0, acc0, false, false);
        acc1 = __builtin_amdgcn_wmma_f32_16x16x4_f32(
            false, a[kt + 1], false, bw[kt + 1], (short)0, acc1, false, false);
    }

    // WMMA consumed all loads (loadcnt drained) before this barrier, so the
    // in-place stores below cannot race other waves' reads of these rows.
    __syncthreads();

#pragma unroll
    for (int r = 0; r < 8; ++r) {
        const int m = r + 8 * hi;
        out[(rb + m) * H + n] = acc0[r] + acc1[r];
    }
}

extern "C" void kernel_launch(void* const* d_in, const int* in_sizes, int n_in,
                              void* d_out, int out_size, void* d_ws, size_t ws_size,
                              hipStream_t stream) {
    const float* wave_input  = (const float*)d_in[0];
    const float* wave_future = (const float*)d_in[1];
    const float* W_ih        = (const float*)d_in[2];
    const float* W_hh        = (const float*)d_in[3];
    const float* b_ih        = (const float*)d_in[4];
    const float* b_hh        = (const float*)d_in[5];
    const float* W_proj      = (const float*)d_in[6];
    const float* b_proj      = (const float*)d_in[7];
    float* out = (float*)d_out;

    lstm_scan_kernel<<<B_TOT / 16, 512, 0, stream>>>(
        wave_input, wave_future, W_ih, W_hh, b_ih, b_hh, out);
    proj_kernel<<<(B_TOT * FUT) / 16, 128, 0, stream>>>(W_proj, b_proj, out);
}